// BoundaryMaxPooling_27384711479957
// MI455X (gfx1250) — compile-verified
//
#include <hip/hip_runtime.h>
#include <cfloat>

// BoundaryMaxPooling for MI455X (gfx1250, wave32).
//
// out[b, ch, t] = max_{j in [lo(t), hi(t))} feature[b, ch, j]
//   ranges derived from batch-0 segments only (reference indexes seg[0]).
//   ch <  256 -> (floor(s0), ceil(s1)) with s1++ when equal
//   ch >= 256 -> (floor(e0), ceil(e1)) with e0-- when equal
//   empty range -> -FLT_MAX (finfo(float32).min)
//
// Memory-bound (~8.3 MB total traffic -> ~0.36 us at 23.3 TB/s). The CDNA5-
// specific path used here is the async global->LDS copy (ASYNCcnt-tracked,
// single instruction, no VGPR round trip) to stage each 126-float row.

namespace {

constexpr int kB  = 16;
constexpr int kC2 = 512;
constexpr int kC  = 256;
constexpr int kT  = 126;

__global__ __launch_bounds__(128) void boundary_maxpool_kernel(
    const float* __restrict__ feature,   // (16, 512, 126)
    const float* __restrict__ segments,  // (16, 126, 4) -- only batch 0 used
    float* __restrict__ out)             // (16, 512, 126)
{
  __shared__ float row[128];

  const int tid = threadIdx.x;
  const int bc  = blockIdx.x;            // b*512 + ch, 0..8191
  const int ch  = bc & (kC2 - 1);

  // ---- Stage this (b, ch) row (126 floats) into LDS via async copy. ----
  {
    // Lanes 126/127 duplicate j = 125 so EXEC stays full (idempotent write).
    const int j = (tid < kT) ? tid : (kT - 1);

    // Generic shared pointer -> LDS byte offset: per the ISA flat-aperture
    // rule, the LDS address is the low 32 bits, so a 32-bit truncation is exact.
    unsigned lds_off = (unsigned)(size_t)(const void*)&row[j];
    unsigned goff    = (unsigned)(j * 4);
    unsigned long long gbase =
        (unsigned long long)(size_t)(const void*)(feature + (size_t)bc * kT);

    // GVS form: LDS-addr VGPR, 32-bit offset VGPR, 64-bit SGPR base.
    asm volatile("global_load_async_to_lds_b32 %0, %1, %2"
                 :
                 : "v"(lds_off), "v"(goff), "s"(gbase)
                 : "memory");
    // Wait for this wave's async transfer; the barrier below then makes all
    // waves' staged data visible to everyone.
    asm volatile("s_wait_asynccnt 0" ::: "memory");
  }
  __syncthreads();

  if (tid < kT) {
    // Batch-0 segment row for time t = tid (16-byte aligned float4).
    const float4 s = reinterpret_cast<const float4*>(segments)[tid];

    const float a0 = fminf(fmaxf(s.x, 0.0f), 125.0f);
    const float a1 = fminf(fmaxf(s.y, 0.0f), 125.0f);
    const float a2 = fminf(fmaxf(s.z, 0.0f), 125.0f);
    const float a3 = fminf(fmaxf(s.w, 0.0f), 125.0f);

    int r0, r1;
    if (ch < kC) {
      r0 = (int)floorf(a0);
      r1 = (int)ceilf(a1);
      if (r0 == r1) r1 += 1;     // s0 == s1 -> s1 + 1
    } else {
      r0 = (int)floorf(a2);
      r1 = (int)ceilf(a3);
      if (r0 == r1) r0 -= 1;     // e0 == e1 -> e0 - 1
    }

    const int lo = (r0 < 0) ? 0 : r0;
    const int hi = (r1 > kT) ? kT : r1;

    float m = -FLT_MAX;          // == jnp.finfo(float32).min
    for (int j = lo; j < hi; ++j) {
      m = fmaxf(m, row[j]);
    }

    out[(size_t)bc * kT + tid] = m;
  }
}

}  // namespace

extern "C" void kernel_launch(void* const* d_in, const int* in_sizes, int n_in,
                              void* d_out, int out_size, void* d_ws, size_t ws_size,
                              hipStream_t stream) {
  const float* feature  = (const float*)d_in[0];  // 16*512*126 fp32
  const float* segments = (const float*)d_in[1];  // 16*126*4 fp32
  float* out = (float*)d_out;                     // 16*512*126 fp32

  (void)in_sizes; (void)n_in; (void)out_size; (void)d_ws; (void)ws_size;

  dim3 grid(kB * kC2);   // 8192 blocks: one (b, ch) row each
  dim3 block(128);       // 4 waves (wave32)
  hipLaunchKernelGGL(boundary_maxpool_kernel, grid, block, 0, stream,
                     feature, segments, out);
}